// FlowNetCorr_27668179321180
// MI455X (gfx1250) — compile-verified
//
#include <hip/hip_runtime.h>
#include <hip/hip_bf16.h>

// ---------------------------------------------------------------------------
// FlowNet-style 3D correlation registration network for MI455X (gfx1250).
// Stride-1 SAME convs run as implicit-GEMM on V_WMMA_F32_16X16X4_F32 (fp32
// matrix path keeps reference numerics). The WMMA inner loop is branchless:
// invalid voxels/rows are clamped (they only feed unused tile rows/columns),
// tap decode uses compile-time constant divisions, addressing is shift/mask.
// Correlation is VALU SAD on voxel-major (contiguous float4) features.
// Working set (~140MB) fits in the 192MB L2.
// ---------------------------------------------------------------------------

typedef float v2f __attribute__((ext_vector_type(2)));
typedef float v8f __attribute__((ext_vector_type(8)));

#define V32 32768   // 32^3
#define V64 262144  // 64^3

static __device__ __forceinline__ float d_lrelu(float v, float a) {
  return v >= 0.f ? v : a * v;
}

// ---------------------------------------------------------------------------
// fe0: conv 1->8, k=7, stride 2, VALID, then pad (2 before, 1 after) per axis.
// Valid padded output coords are [2,30]; border stays exactly 0 (bias added
// pre-pad in the reference). 64^3 input -> 8 x 32^3 output.
// ---------------------------------------------------------------------------
__global__ void k_fe0(const float* __restrict__ img,
                      const float* __restrict__ w,
                      const float* __restrict__ b,
                      float* __restrict__ dst) {
  int idx = blockIdx.x * blockDim.x + threadIdx.x;
  if (idx >= 8 * V32) return;
  int oc = idx >> 15;
  int v  = idx & (V32 - 1);
  int x = v & 31, y = (v >> 5) & 31, z = v >> 10;
  float o = 0.f;
  if (x >= 2 && x < 31 && y >= 2 && y < 31 && z >= 2 && z < 31) {
    int bx = (x - 2) * 2, by = (y - 2) * 2, bz = (z - 2) * 2;
    const float* wp = w + oc * 343;
    float acc = 0.f;
    for (int dz = 0; dz < 7; ++dz)
      for (int dy = 0; dy < 7; ++dy)
        for (int dx = 0; dx < 7; ++dx)
          acc += wp[(dz * 7 + dy) * 7 + dx] *
                 img[((bz + dz) * 64 + (by + dy)) * 64 + (bx + dx)];
    o = acc + b[oc];
  }
  dst[idx] = o;
}

// ---------------------------------------------------------------------------
// BatchNorm training-mode stats: one block per channel, biased variance.
// ---------------------------------------------------------------------------
__global__ void k_bn_stats(const float* __restrict__ src,
                           float* __restrict__ mv, int V) {
  __shared__ float sh_s[256];
  __shared__ float sh_q[256];
  int c = blockIdx.x;
  const float* p = src + (size_t)c * V;
  float s = 0.f, q = 0.f;
  for (int i = threadIdx.x; i < V; i += blockDim.x) {
    float t = p[i];
    s += t;
    q += t * t;
  }
  sh_s[threadIdx.x] = s;
  sh_q[threadIdx.x] = q;
  __syncthreads();
  for (int st = blockDim.x / 2; st > 0; st >>= 1) {
    if ((int)threadIdx.x < st) {
      sh_s[threadIdx.x] += sh_s[threadIdx.x + st];
      sh_q[threadIdx.x] += sh_q[threadIdx.x + st];
    }
    __syncthreads();
  }
  if (threadIdx.x == 0) {
    float m = sh_s[0] / (float)V;
    mv[2 * c]     = m;
    mv[2 * c + 1] = sh_q[0] / (float)V - m * m;
  }
}

__global__ void k_bn_lrelu(float* __restrict__ x,
                           const float* __restrict__ mv,
                           const float* __restrict__ g,
                           const float* __restrict__ beta,
                           int C, int V, float slope) {
  int idx = blockIdx.x * blockDim.x + threadIdx.x;
  if (idx >= C * V) return;
  int c = idx / V;
  float m   = mv[2 * c];
  float var = mv[2 * c + 1];
  float t = (x[idx] - m) * (g[c] * rsqrtf(var + 1e-5f)) + beta[c];
  x[idx] = d_lrelu(t, slope);
}

// ---------------------------------------------------------------------------
// Implicit-GEMM conv3d (SAME, stride 1) on V_WMMA_F32_16X16X4_F32.
// One wave per (16 outch) x (16 contiguous voxels) tile; K = Cin*k^3 in
// steps of 4 (unrolled x2), zero-padded K tail handled in a guarded loop.
// f32 A layout per ISA 7.12.2: lanes 0-15 M=0..15 K0/K1; lanes 16-31 K2/K3.
// B (4x16) mirrored; C/D: lanes 0-15 VGPRr->M=r, lanes 16-31 ->M=r+8.
// Invalid voxels / output rows are clamped, not branched: they only feed
// unused tile columns/rows of D, which are never stored.
// act: 0 = bias only, 1 = bias + LeakyReLU(0.2).
// ---------------------------------------------------------------------------
template <int KSZ>
__global__ void k_conv3d_wmma(const float* __restrict__ src,
                              const float* __restrict__ wgt,
                              const float* __restrict__ bias,
                              float* __restrict__ dst,
                              int Cin, int Cout, int sbits, int act) {
  constexpr int K3  = KSZ * KSZ * KSZ;
  constexpr int KK  = KSZ * KSZ;
  constexpr int OFF = KSZ / 2;
  const int S = 1 << sbits;
  const int V = 1 << (3 * sbits);
  const int lane = threadIdx.x;   // 0..31, one wave per block
  const int half = lane >> 4;     // 0 or 1 (selects K pair)
  const int n    = lane & 15;     // tile column / A row
  const int ocB  = blockIdx.y * 16;
  const int v    = blockIdx.x * 16 + n;
  const bool vok = v < V;
  const int vc   = vok ? v : V - 1;            // clamp: garbage cols unused
  const int x = vc & (S - 1);
  const int y = (vc >> sbits) & (S - 1);
  const int z = vc >> (2 * sbits);
  const int Ktot = Cin * K3;
  const int mrow = min(ocB + n, Cout - 1);     // clamp: garbage rows unused
  const float* wrow = wgt + (size_t)mrow * Ktot;

  // Branchless tap fetch: clamp coords, always load, select zero on OOB.
  auto fetch = [&](int row) -> float {
    int ic = row / K3;
    int rr = row - ic * K3;
    int dz = rr / KK;
    int r2 = rr - dz * KK;
    int dy = r2 / KSZ;
    int dx = r2 - dy * KSZ;
    int iz = z + dz - OFF, iy = y + dy - OFF, ix = x + dx - OFF;
    bool oob = ((unsigned)iz >= (unsigned)S) | ((unsigned)iy >= (unsigned)S) |
               ((unsigned)ix >= (unsigned)S);
    int cz = min(max(iz, 0), S - 1);
    int cy = min(max(iy, 0), S - 1);
    int cx = min(max(ix, 0), S - 1);
    float t = src[(ic << (3 * sbits)) + (((cz << sbits) + cy) << sbits) + cx];
    return oob ? 0.f : t;
  };

  v8f acc = {0.f, 0.f, 0.f, 0.f, 0.f, 0.f, 0.f, 0.f};

  const int Kmain = Ktot & ~7;
  int kk = 0;
  for (; kk < Kmain; kk += 8) {
    int r0 = kk + half * 2;
    v2f a0, b0;
    a0.x = wrow[r0];
    a0.y = wrow[r0 + 1];
    b0.x = fetch(r0);
    b0.y = fetch(r0 + 1);
    acc = __builtin_amdgcn_wmma_f32_16x16x4_f32(false, a0, false, b0,
                                                (short)0, acc, false, false);
    int r4 = r0 + 4;
    v2f a1, b1;
    a1.x = wrow[r4];
    a1.y = wrow[r4 + 1];
    b1.x = fetch(r4);
    b1.y = fetch(r4 + 1);
    acc = __builtin_amdgcn_wmma_f32_16x16x4_f32(false, a1, false, b1,
                                                (short)0, acc, false, false);
  }
  for (; kk < Ktot; kk += 4) {  // K tail: rows >= Ktot must be true zeros
    int r0 = kk + half * 2;
    int r1 = r0 + 1;
    v2f a, b;
    a.x = (r0 < Ktot) ? wrow[r0] : 0.f;
    a.y = (r1 < Ktot) ? wrow[r1] : 0.f;
    b.x = (r0 < Ktot) ? fetch(r0) : 0.f;
    b.y = (r1 < Ktot) ? fetch(r1) : 0.f;
    acc = __builtin_amdgcn_wmma_f32_16x16x4_f32(false, a, false, b,
                                                (short)0, acc, false, false);
  }

  if (vok) {
#pragma unroll
    for (int r = 0; r < 8; ++r) {
      int m  = r + half * 8;
      int oc = ocB + m;
      if (oc < Cout) {
        float o = acc[r] + bias[oc];
        float neg = 0.2f * o;
        if (act == 1) o = o >= 0.f ? o : neg;
        dst[(size_t)oc * V + v] = o;
      }
    }
  }
}

// ---------------------------------------------------------------------------
// Generic small direct conv (stride 1 SAME), out = scale*(conv + bias).
// Used for redir (1x1x1), flow head (3^3, x2 fold), refinement (1x1x1).
// ---------------------------------------------------------------------------
__global__ void k_conv3d_direct(const float* __restrict__ src,
                                const float* __restrict__ wgt,
                                const float* __restrict__ bias,
                                float* __restrict__ dst,
                                int Cin, int Cout, int S, int ksz, float scale) {
  int V = S * S * S;
  int idx = blockIdx.x * blockDim.x + threadIdx.x;
  if (idx >= Cout * V) return;
  int oc = idx / V;
  int v  = idx - oc * V;
  int x = v % S, y = (v / S) % S, z = v / (S * S);
  int off = ksz >> 1;
  int k3 = ksz * ksz * ksz;
  const float* wp = wgt + (size_t)oc * Cin * k3;
  float acc = 0.f;
  for (int ic = 0; ic < Cin; ++ic) {
    for (int dz = 0; dz < ksz; ++dz) {
      int iz = z + dz - off;
      if ((unsigned)iz >= (unsigned)S) continue;
      for (int dy = 0; dy < ksz; ++dy) {
        int iy = y + dy - off;
        if ((unsigned)iy >= (unsigned)S) continue;
        for (int dx = 0; dx < ksz; ++dx) {
          int ix = x + dx - off;
          if ((unsigned)ix >= (unsigned)S) continue;
          acc += wp[(ic * ksz + dz) * ksz * ksz + dy * ksz + dx] *
                 src[ic * V + (iz * S + iy) * S + ix];
        }
      }
    }
  }
  dst[idx] = (acc + bias[oc]) * scale;
}

// ---------------------------------------------------------------------------
// Channel-major -> voxel-major transpose (for contiguous SAD loop).
// ---------------------------------------------------------------------------
__global__ void k_transpose(const float* __restrict__ src,
                            float* __restrict__ dstT, int C, int V) {
  int idx = blockIdx.x * blockDim.x + threadIdx.x;
  if (idx >= C * V) return;
  int c = idx / V;
  int v = idx - c * V;
  dstT[(size_t)v * C + c] = src[idx];
}

__global__ void k_sumabs(const float* __restrict__ fixT,
                         float* __restrict__ sa, int C, int V) {
  int v = blockIdx.x * blockDim.x + threadIdx.x;
  if (v >= V) return;
  const float* p = fixT + (size_t)v * C;
  float s = 0.f;
  for (int c = 0; c < C; ++c) s += fabsf(p[c]);
  sa[v] = s;
}

// ---------------------------------------------------------------------------
// 343-channel windowed SAD correlation at 32^3 over 64 channels.
// Offset channel n -> (d0,d1,d2) = (n%7, (n/7)%7, n/49) - 3 on (z,y,x).
// Out-of-range offsets reduce to sum|fix| (precomputed).
// ---------------------------------------------------------------------------
__global__ void k_corr(const float* __restrict__ fixT,
                       const float* __restrict__ movT,
                       const float* __restrict__ saF,
                       float* __restrict__ corr) {
  int idx = blockIdx.x * blockDim.x + threadIdx.x;
  if (idx >= 343 * V32) return;
  int n = idx >> 15;
  int v = idx & (V32 - 1);
  int d0 = n % 7 - 3;
  int d1 = (n / 7) % 7 - 3;
  int d2 = n / 49 - 3;
  int z = v >> 10, y = (v >> 5) & 31, x = v & 31;
  int mz = z + d0, my = y + d1, mx = x + d2;
  float s;
  if ((unsigned)mz < 32u && (unsigned)my < 32u && (unsigned)mx < 32u) {
    int vm = (mz << 10) | (my << 5) | mx;
    const float4* a = (const float4*)(fixT + (size_t)v * 64);
    const float4* b = (const float4*)(movT + (size_t)vm * 64);
    s = 0.f;
#pragma unroll
    for (int t = 0; t < 16; ++t) {
      float4 fa = a[t], fb = b[t];
      s += fabsf(fa.x - fb.x) + fabsf(fa.y - fb.y) +
           fabsf(fa.z - fb.z) + fabsf(fa.w - fb.w);
    }
  } else {
    s = saF[v];
  }
  corr[idx] = s;
}

// Box filter (k=3, pad 1, count_include_pad) along one axis (extent 32).
__global__ void k_box3(const float* __restrict__ src,
                       float* __restrict__ dst, int stride, int total) {
  int idx = blockIdx.x * blockDim.x + threadIdx.x;
  if (idx >= total) return;
  int pos = (idx / stride) & 31;
  float s = src[idx];
  if (pos > 0)  s += src[idx - stride];
  if (pos < 31) s += src[idx + stride];
  dst[idx] = s * (1.f / 3.f);
}

// 2x2x2 max pool.
__global__ void k_maxpool2(const float* __restrict__ src,
                           float* __restrict__ dst, int C, int S) {
  int So = S >> 1;
  int Vo = So * So * So;
  int idx = blockIdx.x * blockDim.x + threadIdx.x;
  if (idx >= C * Vo) return;
  int c = idx / Vo;
  int v = idx - c * Vo;
  int x = (v % So) * 2, y = ((v / So) % So) * 2, z = (v / (So * So)) * 2;
  const float* p = src + (size_t)c * S * S * S;
  float m = -3.4e38f;
  for (int dz = 0; dz < 2; ++dz)
    for (int dy = 0; dy < 2; ++dy)
      for (int dx = 0; dx < 2; ++dx) {
        float t = p[((z + dz) * S + (y + dy)) * S + (x + dx)];
        m = fmaxf(m, t);
      }
  dst[idx] = m;
}

// Nearest x2 upsample of `a` (C1 @ S/2) concatenated with `skip` (C2 @ S).
__global__ void k_upconcat(const float* __restrict__ a,
                           const float* __restrict__ skip,
                           float* __restrict__ dst,
                           int C1, int C2, int S) {
  int V = S * S * S;
  int Sh = S >> 1;
  int idx = blockIdx.x * blockDim.x + threadIdx.x;
  if (idx >= (C1 + C2) * V) return;
  int c = idx / V;
  int v = idx - c * V;
  if (c < C1) {
    int x = v % S, y = (v / S) % S, z = v / (S * S);
    dst[idx] = a[(size_t)c * Sh * Sh * Sh +
                 ((z >> 1) * Sh + (y >> 1)) * Sh + (x >> 1)];
  } else {
    dst[idx] = skip[(size_t)(c - C1) * V + v];
  }
}

// Trilinear x2 upsample, align_corners=True (separable == direct trilinear).
__global__ void k_tri2x(const float* __restrict__ src,
                        float* __restrict__ dst, int C, int Sin) {
  int Sout = Sin * 2;
  int V = Sout * Sout * Sout;
  int idx = blockIdx.x * blockDim.x + threadIdx.x;
  if (idx >= C * V) return;
  int c = idx / V;
  int v = idx - c * V;
  int x = v % Sout, y = (v / Sout) % Sout, z = v / (Sout * Sout);
  float sc = (float)(Sin - 1) / (float)(Sout - 1);
  float px = x * sc, py = y * sc, pz = z * sc;
  int lx = (int)px, ly = (int)py, lz = (int)pz;
  float fx = px - lx, fy = py - ly, fz = pz - lz;
  int hx = min(lx + 1, Sin - 1), hy = min(ly + 1, Sin - 1), hz = min(lz + 1, Sin - 1);
  const float* p = src + (size_t)c * Sin * Sin * Sin;
  int SS = Sin * Sin;
#define G(Z, Y, X) p[(Z)*SS + (Y)*Sin + (X)]
  float c00 = G(lz, ly, lx) * (1 - fx) + G(lz, ly, hx) * fx;
  float c01 = G(lz, hy, lx) * (1 - fx) + G(lz, hy, hx) * fx;
  float c10 = G(hz, ly, lx) * (1 - fx) + G(hz, ly, hx) * fx;
  float c11 = G(hz, hy, lx) * (1 - fx) + G(hz, hy, hx) * fx;
#undef G
  float c0 = c00 * (1 - fy) + c01 * fy;
  float c1 = c10 * (1 - fy) + c11 * fy;
  dst[idx] = c0 * (1 - fz) + c1 * fz;
}

// ---------------------------------------------------------------------------
// Host-side orchestration.
// ---------------------------------------------------------------------------
enum {
  IN_MOV = 0, IN_FIX,
  FE0_W, FE0_B, FE0_G, FE0_BB,
  FE1_W, FE1_B, FE1_G, FE1_BB,
  FE2_W, FE2_B, FE2_G, FE2_BB,
  RD_W, RD_B, RD_G, RD_BB,
  ENC0_W, ENC0_B, ENC1_W, ENC1_B, ENC2_W, ENC2_B, ENC3_W, ENC3_B,
  DEC0_W, DEC0_B, DEC1_W, DEC1_B, DEC2_W, DEC2_B, DEC3_W, DEC3_B,
  REM0_W, REM0_B, REM1_W, REM1_B, REM2_W, REM2_B,
  FLOW_W, FLOW_B, REF_W, REF_B
};

extern "C" void kernel_launch(void* const* d_in, const int* in_sizes, int n_in,
                              void* d_out, int out_size, void* d_ws, size_t ws_size,
                              hipStream_t stream) {
  (void)in_sizes; (void)n_in; (void)out_size; (void)ws_size;
  auto F = [&](int i) { return (const float*)d_in[i]; };
  float* W = (float*)d_ws;

  // ---- workspace layout (floats) -----------------------------------------
  size_t o = 0;
  float* featM = W + o; o += (size_t)64 * V32;   // moving features 64@32^3
  float* featF = W + o; o += (size_t)64 * V32;   // fixed features
  float* bufA  = W + o; o += (size_t)32 * V32;   // fe scratch
  float* bufB  = W + o; o += (size_t)32 * V32;   // fe scratch (contiguous)
  float* featMT = bufA;                          // alias: mov voxel-major (needs 64*V32)
  float* corr  = W + o; o += (size_t)343 * V32;
  float* xbuf  = W + o; o += (size_t)375 * V32;  // corr||redir, also box ping
  float* e0 = W + o; o += (size_t)16 * V32;
  float* e1 = W + o; o += (size_t)32 * 4096;
  float* e2 = W + o; o += (size_t)32 * 512;
  float* e3 = W + o; o += (size_t)32 * 64;
  float* ua = W + o; o += (size_t)48 * V32;      // unet ping
  float* ub = W + o; o += (size_t)48 * V32;      // unet pong
  float* featFT = ua;                            // alias: fix voxel-major (done pre-unet)
  float* sumabs = W + o; o += V32;
  float* flowb  = W + o; o += (size_t)3 * V32;
  float* flowup = W + o; o += (size_t)3 * V64;
  float* bn     = W + o; o += 128;               // mean/var pairs, <=64 ch

  auto cdiv = [](long a, long b) { return (unsigned)((a + b - 1) / b); };
  const int T = 256;

  // ---- feature extractor (per image: conv+BN+LReLU 0.01 x3) --------------
  auto run_fe = [&](const float* img, float* feat) {
    k_fe0<<<cdiv(8L * V32, T), T, 0, stream>>>(img, F(FE0_W), F(FE0_B), bufA);
    k_bn_stats<<<8, T, 0, stream>>>(bufA, bn, V32);
    k_bn_lrelu<<<cdiv(8L * V32, T), T, 0, stream>>>(bufA, bn, F(FE0_G), F(FE0_BB), 8, V32, 0.01f);

    dim3 g1(cdiv(V32, 16), 2);  // Cout=32
    k_conv3d_wmma<5><<<g1, 32, 0, stream>>>(bufA, F(FE1_W), F(FE1_B), bufB, 8, 32, 5, 0);
    k_bn_stats<<<32, T, 0, stream>>>(bufB, bn, V32);
    k_bn_lrelu<<<cdiv(32L * V32, T), T, 0, stream>>>(bufB, bn, F(FE1_G), F(FE1_BB), 32, V32, 0.01f);

    dim3 g2(cdiv(V32, 16), 4);  // Cout=64
    k_conv3d_wmma<5><<<g2, 32, 0, stream>>>(bufB, F(FE2_W), F(FE2_B), feat, 32, 64, 5, 0);
    k_bn_stats<<<64, T, 0, stream>>>(feat, bn, V32);
    k_bn_lrelu<<<cdiv(64L * V32, T), T, 0, stream>>>(feat, bn, F(FE2_G), F(FE2_BB), 64, V32, 0.01f);
  };
  run_fe(F(IN_MOV), featM);
  run_fe(F(IN_FIX), featF);

  // ---- correlation --------------------------------------------------------
  k_transpose<<<cdiv(64L * V32, T), T, 0, stream>>>(featM, featMT, 64, V32);
  k_transpose<<<cdiv(64L * V32, T), T, 0, stream>>>(featF, featFT, 64, V32);
  k_sumabs<<<cdiv(V32, T), T, 0, stream>>>(featFT, sumabs, 64, V32);
  k_corr<<<cdiv(343L * V32, T), T, 0, stream>>>(featFT, featMT, sumabs, corr);

  // 3-axis box filter: corr -> xbuf -> corr -> xbuf[0:343*V32]
  const int NC = 343 * V32;
  k_box3<<<cdiv(NC, T), T, 0, stream>>>(corr, xbuf, 1024, NC);
  k_box3<<<cdiv(NC, T), T, 0, stream>>>(xbuf, corr, 32, NC);
  k_box3<<<cdiv(NC, T), T, 0, stream>>>(corr, xbuf, 1, NC);

  // ---- redir: 1x1x1 conv 64->32 on moving features + BN + LReLU 0.01 -----
  float* redir = xbuf + (size_t)343 * V32;
  k_conv3d_direct<<<cdiv(32L * V32, T), T, 0, stream>>>(featM, F(RD_W), F(RD_B),
                                                        redir, 64, 32, 32, 1, 1.f);
  k_bn_stats<<<32, T, 0, stream>>>(redir, bn, V32);
  k_bn_lrelu<<<cdiv(32L * V32, T), T, 0, stream>>>(redir, bn, F(RD_G), F(RD_BB), 32, V32, 0.01f);

  // ---- U-Net --------------------------------------------------------------
  auto conv = [&](const float* s, int wi, int bi, float* d,
                  int Cin, int Cout, int sbits) {
    long V = 1L << (3 * sbits);
    dim3 g(cdiv(V, 16), cdiv(Cout, 16));
    k_conv3d_wmma<3><<<g, 32, 0, stream>>>(s, F(wi), F(bi), d, Cin, Cout, sbits, 1);
  };
  auto pool = [&](const float* s, float* d, int C, int S) {
    k_maxpool2<<<cdiv((long)C * (S / 2) * (S / 2) * (S / 2), T), T, 0, stream>>>(s, d, C, S);
  };
  auto upcat = [&](const float* a, const float* sk, float* d, int C1, int C2, int S) {
    k_upconcat<<<cdiv((long)(C1 + C2) * S * S * S, T), T, 0, stream>>>(a, sk, d, C1, C2, S);
  };

  // encoder (sbits: 32^3 -> 5, 16^3 -> 4, 8^3 -> 3, 4^3 -> 2, 2^3 -> 1)
  conv(xbuf, ENC0_W, ENC0_B, e0, 375, 16, 5);
  pool(e0, ua, 16, 32);                 // 16 @ 16^3
  conv(ua, ENC1_W, ENC1_B, e1, 16, 32, 4);
  pool(e1, ua, 32, 16);                 // 32 @ 8^3
  conv(ua, ENC2_W, ENC2_B, e2, 32, 32, 3);
  pool(e2, ua, 32, 8);                  // 32 @ 4^3
  conv(ua, ENC3_W, ENC3_B, e3, 32, 32, 2);
  pool(e3, ua, 32, 4);                  // 32 @ 2^3

  // decoder
  conv(ua, DEC0_W, DEC0_B, ub, 32, 32, 1);
  upcat(ub, e3, ua, 32, 32, 4);         // 64 @ 4^3
  conv(ua, DEC1_W, DEC1_B, ub, 64, 32, 2);
  upcat(ub, e2, ua, 32, 32, 8);         // 64 @ 8^3
  conv(ua, DEC2_W, DEC2_B, ub, 64, 32, 3);
  upcat(ub, e1, ua, 32, 32, 16);        // 64 @ 16^3
  conv(ua, DEC3_W, DEC3_B, ub, 64, 32, 4);
  upcat(ub, e0, ua, 32, 16, 32);        // 48 @ 32^3

  // remainder
  conv(ua, REM0_W, REM0_B, ub, 48, 32, 5);
  conv(ub, REM1_W, REM1_B, ua, 32, 16, 5);
  conv(ua, REM2_W, REM2_B, ub, 16, 16, 5);

  // ---- flow head, x2 fold, trilinear x2, refinement -----------------------
  k_conv3d_direct<<<cdiv(3L * V32, T), T, 0, stream>>>(ub, F(FLOW_W), F(FLOW_B),
                                                       flowb, 16, 3, 32, 3, 2.0f);
  k_tri2x<<<cdiv(3L * V64, T), T, 0, stream>>>(flowb, flowup, 3, 32);
  k_conv3d_direct<<<cdiv(3L * V64, T), T, 0, stream>>>(flowup, F(REF_W), F(REF_B),
                                                       (float*)d_out, 3, 3, 64, 1, 1.f);
}